// NormalMixtureEM_88081189306555
// MI455X (gfx1250) — compile-verified
//
#include <hip/hip_runtime.h>

typedef __attribute__((ext_vector_type(2))) float v2f;
typedef __attribute__((ext_vector_type(8))) float v8f;

#define EPSF 1e-8f
constexpr int BN = 4096;   // batch rows
constexpr int LN = 1024;   // window length
constexpr int KC = 4;      // mixture components
constexpr int NITER = 5;
constexpr int PSTRIDE = 20;  // LDS row stride in floats (80B: b128-aligned, low conflicts)
constexpr float LOG2E  = 1.4426950408889634f;
constexpr float HALF_LOG2E = 0.7213475204444817f;  // 0.5/ln(2)

// Block-wide sum of N (<=12) floats across 256 threads (8 wave32s), done almost
// entirely on the matrix pipe:
//   1) each thread writes its padded partials as row `tid` of P[256][20]
//   2) each wave sums its own 32 rows with 2 independent chains of 4
//      V_WMMA_F32_16X16X4_F32 (A = all-ones => D[.,n] = sum over the 4 B-rows of
//      column n; invariant to fragment row permutation, so only the documented
//      column<->lane mapping N = lane&15 matters)
//   3) the 8 wave-total rows are combined by 2 more WMMAs
//   4) totals broadcast to every lane through a 16-float LDS line
template <int N>
__device__ __forceinline__ void block_reduce(float* v, int tid, int wave,
                                             int lane, float (*P)[PSTRIDE],
                                             float (*Q)[PSTRIDE], float* T) {
  constexpr int M4 = (N + 3) / 4;  // caller pads v[] to 4*M4 entries
#pragma unroll
  for (int c = 0; c < M4; ++c)
    *(float4*)&P[tid][4 * c] =
        make_float4(v[4 * c], v[4 * c + 1], v[4 * c + 2], v[4 * c + 3]);
  __syncthreads();

  const int half = lane >> 4;
  const int n = lane & 15;
  v2f A;
  A[0] = 1.0f;
  A[1] = 1.0f;
  const int base = wave * 32 + 2 * half;

  v8f acca = {}, accb = {};
#pragma unroll
  for (int i = 0; i < 4; ++i) {  // rows base.. (+4 per step), 2 rows per lane
    v2f Bf;
    Bf[0] = P[base + 4 * i][n];
    Bf[1] = P[base + 4 * i + 1][n];
    acca = __builtin_amdgcn_wmma_f32_16x16x4_f32(false, A, false, Bf, (short)0,
                                                 acca, false, false);
  }
#pragma unroll
  for (int i = 4; i < 8; ++i) {
    v2f Bf;
    Bf[0] = P[base + 4 * i][n];
    Bf[1] = P[base + 4 * i + 1][n];
    accb = __builtin_amdgcn_wmma_f32_16x16x4_f32(false, A, false, Bf, (short)0,
                                                 accb, false, false);
  }
  if (lane < N) Q[wave][lane] = acca[0] + accb[0];  // lane n holds wave total[n]
  __syncthreads();

  v2f C0, C1;
  C0[0] = Q[2 * half + 0][n];
  C0[1] = Q[2 * half + 1][n];
  C1[0] = Q[2 * half + 4][n];
  C1[1] = Q[2 * half + 5][n];
  v8f f0 = {}, f1 = {};
  f0 = __builtin_amdgcn_wmma_f32_16x16x4_f32(false, A, false, C0, (short)0, f0,
                                             false, false);
  f1 = __builtin_amdgcn_wmma_f32_16x16x4_f32(false, A, false, C1, (short)0, f1,
                                             false, false);
  if (lane < N) T[lane] = f0[0] + f1[0];  // grand total[n]
  __syncthreads();
#pragma unroll
  for (int c = 0; c < M4; ++c) {
    float4 t4 = *(const float4*)&T[4 * c];
    v[4 * c] = t4.x;
    v[4 * c + 1] = t4.y;
    v[4 * c + 2] = t4.z;
    v[4 * c + 3] = t4.w;
  }
}

__global__ __launch_bounds__(256) void NormalMixtureEM_kernel(
    const float* __restrict__ windows,        // [B,L]
    const float* __restrict__ noise,          // [B,K]
    const float* __restrict__ centers,        // [K]
    const float* __restrict__ scales,         // [K]
    const float* __restrict__ weights,        // [K]
    const float* __restrict__ prior_p_param,  // [1]
    const float* __restrict__ mix_weights,    // [1,L]
    const float* __restrict__ blend,          // [1]
    float* __restrict__ out)                  // g | p | a | b
{
  __shared__ float P[256][PSTRIDE];  // per-thread partial rows (20 KB)
  __shared__ float Q[8][PSTRIDE];    // per-wave totals
  __shared__ float T[16];            // broadcast line
  const int tid = threadIdx.x;
  const int b = blockIdx.x;
  const int wave = tid >> 5;
  const int lane = tid & 31;

  // Row data: 4 contiguous elements per lane (coalesced b128 loads).
  const float4 x4 = *(const float4*)(windows + (size_t)b * LN + tid * 4);
  const float4 m4 = *(const float4*)(mix_weights + tid * 4);
  float xx[4] = {x4.x, x4.y, x4.z, x4.w};
  float ww[4] = {__builtin_amdgcn_exp2f(m4.x * LOG2E),
                 __builtin_amdgcn_exp2f(m4.y * LOG2E),
                 __builtin_amdgcn_exp2f(m4.z * LOG2E),
                 __builtin_amdgcn_exp2f(m4.w * LOG2E)};

  // ---- row statistics (unbiased) ----
  float red[4];
  red[0] = (xx[0] + xx[1]) + (xx[2] + xx[3]);
  red[1] = (xx[0] * xx[0] + xx[1] * xx[1]) + (xx[2] * xx[2] + xx[3] * xx[3]);
  red[2] = 0.0f;
  red[3] = 0.0f;
  block_reduce<2>(red, tid, wave, lane, P, Q, T);
  const float mean = red[0] * (1.0f / LN);
  const float varU = fmaxf((red[1] - red[0] * mean) * (1.0f / (LN - 1)), 0.0f);
  const float stdU = __builtin_amdgcn_sqrtf(varU);

  const float prior_p =
      __builtin_amdgcn_rcpf(1.0f + __builtin_amdgcn_exp2f(-prior_p_param[0] * LOG2E));
  const float blender =
      __builtin_amdgcn_rcpf(1.0f + __builtin_amdgcn_exp2f(-blend[0] * LOG2E));
  const float omb = 1.0f - blender;
  const float prior_a = mean;
  const float prior_var = varU;

  // ---- initialize parameters ----
  float a_[KC], b_[KC], p_[KC];
#pragma unroll
  for (int k = 0; k < KC; ++k) {
    a_[k] = centers[k] * stdU + mean + noise[(size_t)b * KC + k] * stdU * 0.01f;
    b_[k] = fabsf(scales[k]) * stdU;
    p_[k] = weights[k];
  }

  for (int it = 0; it < NITER; ++it) {
    float rb[KC], lw2[KC];
#pragma unroll
    for (int k = 0; k < KC; ++k) {
      rb[k] = __builtin_amdgcn_rcpf(b_[k] + EPSF);             // 1/(b+eps)
      lw2[k] = __builtin_amdgcn_logf((p_[k] + EPSF) * rb[k]);  // log2 weights
    }

    // ---- e-step + weighted moments s0=sum(gw), s1=sum(gw*x), s2=sum(gw*x^2)
    float acc[12] = {0, 0, 0, 0, 0, 0, 0, 0, 0, 0, 0, 0};
#pragma unroll
    for (int j = 0; j < 4; ++j) {
      float lm[KC];
#pragma unroll
      for (int k = 0; k < KC; ++k) {
        float z = (xx[j] - a_[k]) * rb[k];
        lm[k] = __fmaf_rn(z * z, -HALF_LOG2E, lw2[k]);  // log2-domain logit
      }
      float mx = fmaxf(fmaxf(lm[0], lm[1]), fmaxf(lm[2], lm[3]));
      float e[KC], s = 0.0f;
#pragma unroll
      for (int k = 0; k < KC; ++k) {
        e[k] = __builtin_amdgcn_exp2f(lm[k] - mx);
        s += e[k];
      }
      float inv = __builtin_amdgcn_rcpf(s);
      if (it == NITER - 1) {  // g of the final e-step is an output
        float4 go = make_float4(e[0] * inv, e[1] * inv, e[2] * inv, e[3] * inv);
        *(float4*)(out + ((size_t)b * LN + (size_t)tid * 4 + j) * KC) = go;
      }
#pragma unroll
      for (int k = 0; k < KC; ++k) {
        float g = e[k] * inv * ww[j];
        float gx = g * xx[j];
        acc[k] += g;
        acc[4 + k] += gx;
        acc[8 + k] += gx * xx[j];
      }
    }
    block_reduce<12>(acc, tid, wave, lane, P, Q, T);

    // ---- m-step (variance via expanded second moment) ----
    float psum = 0.0f, praw[KC], rsg[KC], an[KC];
#pragma unroll
    for (int k = 0; k < KC; ++k) {
      float sg = fmaxf(acc[k], EPSF);
      rsg[k] = __builtin_amdgcn_rcpf(sg);
      praw[k] = sg + prior_p;
      psum += praw[k];
      an[k] = (acc[4 + k] * rsg[k]) * blender + prior_a * omb;
    }
    float pinv = __builtin_amdgcn_rcpf(psum);
#pragma unroll
    for (int k = 0; k < KC; ++k) {
      p_[k] = praw[k] * pinv;
      // sum gw*(x-a)^2 = s2 - 2a*s1 + a^2*s0
      float vnum = __fmaf_rn(
          an[k], __fmaf_rn(an[k], acc[k], -2.0f * acc[4 + k]), acc[8 + k]);
      float dv = fmaxf(vnum, 0.0f) * rsg[k];
      b_[k] = __builtin_amdgcn_sqrtf(dv * blender + prior_var * omb + EPSF);
      a_[k] = an[k];
    }
  }

  // ---- write p, a, b (all lanes hold all K values; one lane stores) ----
  if (tid == 0) {
    float* base = out + (size_t)BN * LN * KC;
    *(float4*)(base + (size_t)b * KC) = make_float4(p_[0], p_[1], p_[2], p_[3]);
    *(float4*)(base + (size_t)BN * KC + (size_t)b * KC) =
        make_float4(a_[0], a_[1], a_[2], a_[3]);
    *(float4*)(base + (size_t)2 * BN * KC + (size_t)b * KC) =
        make_float4(b_[0], b_[1], b_[2], b_[3]);
  }
}

extern "C" void kernel_launch(void* const* d_in, const int* in_sizes, int n_in,
                              void* d_out, int out_size, void* d_ws,
                              size_t ws_size, hipStream_t stream) {
  (void)in_sizes; (void)n_in; (void)d_ws; (void)ws_size; (void)out_size;
  NormalMixtureEM_kernel<<<BN, 256, 0, stream>>>(
      (const float*)d_in[0],  // windows
      (const float*)d_in[1],  // noise
      (const float*)d_in[2],  // init_centers
      (const float*)d_in[3],  // init_scales
      (const float*)d_in[4],  // init_weights
      (const float*)d_in[5],  // prior_p_param
      (const float*)d_in[6],  // mix_weights
      (const float*)d_in[7],  // blend
      (float*)d_out);
}